// BinaryTensor_52913997086766
// MI455X (gfx1250) — compile-verified
//
#include <hip/hip_runtime.h>
#include <stdint.h>

// ---------------------------------------------------------------------------
// BinaryTensor vote/update for MI455X (gfx1250).
//
// Memory-bound packed-bit problem (~25 MB total traffic ~= 1.1 us @ 23.3TB/s).
// WMMA intentionally NOT used: any matmul formulation requires unpacking bits
// to bytes (8x traffic). Instead: wave32 bit-sliced Harley-Seal CSA popcount
// (~150 VALU ops per 32 bit-lanes), 6-bit bit-plane intermediate (3 MB), and
// gfx1250 async-to-LDS double-buffered streaming of the flip tensor.
// ---------------------------------------------------------------------------

#define NPOS      131072        // uint32 positions per vote slice (2048*256/4)
#define NBYTES_W  524288        // weights bytes == output weight floats
#define TOTBITS_F 4194304.0f    // 2048*2048 bits

#if defined(__HIP_DEVICE_COMPILE__) && defined(__gfx1250__)
#if __has_builtin(__builtin_amdgcn_global_load_async_to_lds_b32) && \
    __has_builtin(__builtin_amdgcn_s_wait_asynccnt)
#define USE_ASYNC_LDS 1
#endif
#endif

#ifdef USE_ASYNC_LDS
// Builtin signature (from clang diagnostic): param 1 is addrspace(1) int*
// (prints as "__device__ int *" in HIP), param 2 is addrspace(3) int*.
// Global generic ptr bits == addrspace(1) ptr bits; generic LDS address low
// 32 bits == LDS byte offset (flat aperture rule), so int->ptr casts are exact.
#define AS1_INT(x) ((__attribute__((address_space(1))) int*)(uintptr_t)(x))
#define AS3_INT(x) ((__attribute__((address_space(3))) int*)(uint32_t)(uintptr_t)(x))
#endif

// Carry-save adder: per bit-lane, a+b+c = lo + 2*hi (5 VALU ops / 32 lanes).
__device__ __forceinline__ void csa(uint32_t& hi, uint32_t& lo,
                                    uint32_t a, uint32_t b, uint32_t c) {
  uint32_t u = a ^ b;
  hi = (a & b) | (u & c);
  lo = u ^ c;
}

__device__ __forceinline__ unsigned block_reduce_add(unsigned v) {
  __shared__ unsigned red[256];
  red[threadIdx.x] = v;
  __syncthreads();
#pragma unroll
  for (int s = 128; s > 0; s >>= 1) {
    if ((int)threadIdx.x < s) red[threadIdx.x] += red[threadIdx.x + s];
    __syncthreads();
  }
  return red[0];
}

__global__ void zero_counters_kernel(unsigned* __restrict__ counters) {
  counters[0] = 0u;  // total vote popcount (mean numerator)
  counters[1] = 0u;  // flip_mask popcount (update_ratio numerator)
}

// Pass 1: bit-sliced vote counts. Each thread owns one uint32 lane-group
// (32 bit positions) and sums that word across the 32 vote slices into six
// bit-planes s0..s5 (counts 0..32 fit in 6 bits). Planes -> ws (plane-major),
// total popcount -> counters[0].
__global__ void __launch_bounds__(256) votes_kernel(
    const uint32_t* __restrict__ flipw,
    uint32_t* __restrict__ planes,
    unsigned* __restrict__ counters)
{
  const int tid = threadIdx.x;
  const int p   = blockIdx.x * 256 + tid;

  uint32_t ones = 0, twos = 0, fours = 0;
  uint32_t eights[4];
  uint32_t w[8];

#ifdef USE_ASYNC_LDS
  __shared__ uint32_t sbuf[2][8][256];  // 16 KB double buffer
  {
    const uint32_t* g0 = flipw + p;
#pragma unroll
    for (int j = 0; j < 8; ++j)
      __builtin_amdgcn_global_load_async_to_lds_b32(
          AS1_INT(g0 + j * NPOS), AS3_INT(&sbuf[0][j][tid]), 0, 0);
  }
#endif

#pragma unroll
  for (int g = 0; g < 4; ++g) {
#ifdef USE_ASYNC_LDS
    if (g < 3) {
      // Prefetch next group of 8 vote words while group g completes.
      const uint32_t* gn = flipw + (g + 1) * 8 * NPOS + p;
#pragma unroll
      for (int j = 0; j < 8; ++j)
        __builtin_amdgcn_global_load_async_to_lds_b32(
            AS1_INT(gn + j * NPOS), AS3_INT(&sbuf[(g + 1) & 1][j][tid]), 0, 0);
      __builtin_amdgcn_s_wait_asynccnt(8);  // group g's 8 loads retired (in-order)
    } else {
      __builtin_amdgcn_s_wait_asynccnt(0);
    }
#pragma unroll
    for (int j = 0; j < 8; ++j) w[j] = sbuf[g & 1][j][tid];
#else
    const uint32_t* gp = flipw + g * 8 * NPOS + p;
#pragma unroll
    for (int j = 0; j < 8; ++j) w[j] = gp[j * NPOS];
#endif
    // Harley-Seal: 8 words -> ones/twos/fours accumulators + one eights word.
    uint32_t tA, tB, fA, fB;
    csa(tA, ones, ones, w[0], w[1]);
    csa(tB, ones, ones, w[2], w[3]);
    csa(fA, twos, twos, tA, tB);
    csa(tA, ones, ones, w[4], w[5]);
    csa(tB, ones, ones, w[6], w[7]);
    csa(fB, twos, twos, tA, tB);
    csa(eights[g], fours, fours, fA, fB);
  }

  // Fold the four eights words: e0+e1+e2+e3 = s8 + 2*s16 + 4*c32 (weights 8/16/32).
  uint32_t s8, c16a;
  csa(c16a, s8, eights[0], eights[1], eights[2]);
  uint32_t c16b = s8 & eights[3];
  s8 ^= eights[3];
  uint32_t s16 = c16a ^ c16b;
  uint32_t c32 = c16a & c16b;

  planes[0 * NPOS + p] = ones;
  planes[1 * NPOS + p] = twos;
  planes[2 * NPOS + p] = fours;
  planes[3 * NPOS + p] = s8;
  planes[4 * NPOS + p] = s16;
  planes[5 * NPOS + p] = c32;

  unsigned cnt = (unsigned)__popc(ones)
               + ((unsigned)__popc(twos)  << 1)
               + ((unsigned)__popc(fours) << 2)
               + ((unsigned)__popc(s8)    << 3)
               + ((unsigned)__popc(s16)   << 4)
               + ((unsigned)__popc(c32)   << 5);
  unsigned total = block_reduce_add(cnt);
  if (tid == 0) atomicAdd(&counters[0], total);
}

// Pass 2: threshold, XNOR update, ratio count.
// votes > t  <=>  votes >= floor(t)+1 = K  (votes integral, t >= 0).
__global__ void __launch_bounds__(256) apply_kernel(
    const uint32_t* __restrict__ weightsw,
    const uint32_t* __restrict__ planes,
    const unsigned* __restrict__ counters,
    const int*      __restrict__ n_votes_p,
    const float*    __restrict__ vote_p_max_p,
    float*          __restrict__ out,
    unsigned*       __restrict__ flip_counter)
{
  const int p = blockIdx.x * 256 + threadIdx.x;

  const float nv     = (float)n_votes_p[0];
  const float mean   = (float)counters[0] / TOTBITS_F;           // == votes.mean()
  const float thresh = fmaxf(vote_p_max_p[0] * nv, mean);        // p * n_votes
  const uint32_t K   = (uint32_t)((int)floorf(thresh) + 1);      // <= 33, 6 bits

  uint32_t s[6];
#pragma unroll
  for (int i = 0; i < 6; ++i) s[i] = planes[i * NPOS + p];

  // Branchless bit-sliced comparator: mask lane = (count >= K), MSB-down.
  uint32_t gt = 0u, eq = 0xFFFFFFFFu;
#pragma unroll
  for (int i = 5; i >= 0; --i) {
    uint32_t kb = 0u - ((K >> i) & 1u);   // all-ones iff K bit set
    gt |= eq & s[i] & ~kb;                // count bit 1 where K bit 0 -> greater
    eq &= s[i] ^ ~kb;                     // stay equal only on matching bit
  }
  const uint32_t mask = gt | eq;

  // XNOR with weights; same physical bit layout as packbits(big-endian).
  const uint32_t nw = ~(weightsw[p] ^ mask);

  float4 o;
  o.x = (float)( nw        & 0xffu);
  o.y = (float)((nw >> 8)  & 0xffu);
  o.z = (float)((nw >> 16) & 0xffu);
  o.w = (float)( nw >> 24);
  reinterpret_cast<float4*>(out)[p] = o;

  unsigned total = block_reduce_add((unsigned)__popc(mask));
  if (threadIdx.x == 0) atomicAdd(flip_counter, total);
}

__global__ void ratio_kernel(const unsigned* __restrict__ counters,
                             float* __restrict__ out) {
  out[NBYTES_W] = (float)counters[1] / TOTBITS_F;
}

extern "C" void kernel_launch(void* const* d_in, const int* in_sizes, int n_in,
                              void* d_out, int out_size, void* d_ws, size_t ws_size,
                              hipStream_t stream) {
  (void)in_sizes; (void)n_in; (void)out_size; (void)ws_size;

  const uint32_t* weightsw   = (const uint32_t*)d_in[0];  // [2048,256] u8 as u32
  const uint32_t* flipw      = (const uint32_t*)d_in[1];  // [32,2048,256] u8 as u32
  const int*      n_votes    = (const int*)d_in[2];
  const float*    vote_p_max = (const float*)d_in[3];
  float*          out        = (float*)d_out;             // 524288 weight floats + ratio

  unsigned* counters = (unsigned*)d_ws;                       // 2 x u32
  uint32_t* planes   = (uint32_t*)((char*)d_ws + 256);        // 6 x 131072 u32 (3 MB)

  zero_counters_kernel<<<1, 1, 0, stream>>>(counters);
  votes_kernel<<<NPOS / 256, 256, 0, stream>>>(flipw, planes, counters);
  apply_kernel<<<NPOS / 256, 256, 0, stream>>>(weightsw, planes, counters,
                                               n_votes, vote_p_max, out,
                                               counters + 1);
  ratio_kernel<<<1, 1, 0, stream>>>(counters, out);
}